// TopologicalGraphDistillationLoss_56573309223819
// MI455X (gfx1250) — compile-verified
//
#include <hip/hip_runtime.h>
#include <hip/hip_bf16.h>

// ---------------------------------------------------------------------------
// Problem constants (match reference setup_inputs: N=4096 points, d=128 dims)
// ---------------------------------------------------------------------------
constexpr int kN = 4096;   // number of points
constexpr int kD = 128;    // feature dim
constexpr float kBIG = 1e9f;
constexpr float kEPS = 1e-12f;
constexpr int kRounds = 12;           // ceil(log2(kN)) Boruvka rounds
constexpr unsigned long long kSent = ~0ull;

typedef __attribute__((ext_vector_type(2))) float v2f;
typedef __attribute__((ext_vector_type(4))) float v4f;
typedef __attribute__((ext_vector_type(8))) float v8f;

// Workspace layout (float slots; all sub-arrays 8-byte aligned):
constexpr size_t kNN        = (size_t)kN * kN;
constexpr size_t kOffD      = 0;                    // 2 * kN*kN floats (128 MB)
constexpr size_t kOffSq     = 2 * kNN;              // 2 * kN floats
constexpr size_t kOffDeaths = kOffSq + 2 * kN;      // 2 * kN floats
constexpr size_t kOffComp   = kOffDeaths + 2 * kN;  // 2 * kN ints
constexpr size_t kOffParent = kOffComp + 2 * kN;    // 2 * kN ints
constexpr size_t kOffCbest  = kOffParent + 2 * kN;  // 2 * kN u64 (4*kN slots)
constexpr size_t kOffCnt    = kOffCbest + 4 * kN;   // 2 ints
constexpr size_t kOffAccum  = kOffCnt + 2;          // 1 float

__device__ __forceinline__ unsigned long long shfl_xor_u64(unsigned long long v, int mask) {
    unsigned int lo = (unsigned int)v;
    unsigned int hi = (unsigned int)(v >> 32);
    lo = __shfl_xor(lo, mask, 32);
    hi = __shfl_xor(hi, mask, 32);
    return ((unsigned long long)hi << 32) | lo;
}

// pack positive-float weight (order-preserving bits) with (i,j) tie-break
__device__ __forceinline__ unsigned long long pack_key(float w, int i, int j) {
    unsigned int wb = __float_as_uint(w);   // w > 0 always (sqrt of >= EPS)
    return ((unsigned long long)wb << 32) | (unsigned long long)((i << 12) | j);
}

// ---------------------------------------------------------------------------
// 0) init: component labels, parents, best-edge sentinels, counters, accum
// ---------------------------------------------------------------------------
__global__ __launch_bounds__(256) void init_kernel(float* __restrict__ ws) {
    int id = blockIdx.x * blockDim.x + threadIdx.x;   // 0 .. 2*kN-1
    int m = id >> 12;
    int r = id & (kN - 1);
    ((int*)(ws + kOffComp))[id]   = r;
    ((int*)(ws + kOffParent))[id] = r;
    ((unsigned long long*)(ws + kOffCbest))[id] = kSent;
    ws[kOffDeaths + id] = 0.0f;
    if (r == 0) ((int*)(ws + kOffCnt))[m] = 0;
    if (id == 0) ws[kOffAccum] = 0.0f;
}

// ---------------------------------------------------------------------------
// 1) per-row squared norms for both matrices: 8192 rows total
// ---------------------------------------------------------------------------
__global__ __launch_bounds__(256) void sqnorm_kernel(
    const float* __restrict__ Xs, const float* __restrict__ Xt,
    float* __restrict__ ws) {
    int id = blockIdx.x * blockDim.x + threadIdx.x;   // 0 .. 2*kN-1
    int m = id >> 12;
    int r = id & (kN - 1);
    const float* row = (m ? Xt : Xs) + (size_t)r * kD;
    float s = 0.0f;
    #pragma unroll 8
    for (int k = 0; k < kD; ++k) { float v = row[k]; s += v * v; }
    ws[kOffSq + (size_t)m * kN + r] = s;
}

// ---------------------------------------------------------------------------
// 2) WMMA fp32 GEMM (G = X X^T) fused with distance epilogue:
//    D[i][j] = sqrt(max(sq[i]+sq[j]-2*G[i][j], EPS))
//    One wave per 16x16 tile, V_WMMA_F32_16X16X4_F32, K=128 in 32 steps.
// ---------------------------------------------------------------------------
__global__ __launch_bounds__(256) void gemm_dist_kernel(
    const float* __restrict__ Xs, const float* __restrict__ Xt,
    float* __restrict__ ws) {
    const int matrix = blockIdx.y;                 // 0 student / 1 teacher
    const float* X   = matrix ? Xt : Xs;
    const float* sq  = ws + kOffSq + (size_t)matrix * kN;
    float* Dmat      = ws + kOffD + (size_t)matrix * kNN;

    const int lane = threadIdx.x & 31;
    const int wid  = threadIdx.x >> 5;
    const int wg   = blockIdx.x * 8 + wid;         // global wave id
    const int mi   = wg >> 8;                      // tile row (256 tiles/side)
    const int ni   = wg & 255;                     // tile col

    // A layout (16x4 f32): lanes 0-15 hold K={0,1}, lanes 16-31 hold K={2,3}
    const int half = lane >> 4;                    // 0 or 1
    const int l16  = lane & 15;
    const int koff = half * 2;

    const float* arow = X + (size_t)(mi * 16 + l16) * kD + koff;
    const float* brow = X + (size_t)(ni * 16 + l16) * kD + koff;

    v8f c = {};
    #pragma unroll 8
    for (int k = 0; k < kD; k += 4) {
        v2f a = *(const v2f*)(arow + k);
        v2f b = *(const v2f*)(brow + k);
        c = __builtin_amdgcn_wmma_f32_16x16x4_f32(
                /*neg_a=*/false, a, /*neg_b=*/false, b,
                /*c_mod=*/(short)0, c, /*reuse_a=*/false, /*reuse_b=*/false);
    }

    // C/D layout: lane l, VGPR v holds element (M = v + 8*(l>>4), N = l&15)
    const int col  = ni * 16 + l16;
    const float sc = sq[col];
    #pragma unroll
    for (int v = 0; v < 8; ++v) {
        int row = mi * 16 + v + 8 * half;
        float d2 = sq[row] + sc - 2.0f * c[v];
        Dmat[(size_t)row * kN + col] = sqrtf(fmaxf(d2, kEPS));
    }
}

// ---------------------------------------------------------------------------
// 3a) Boruvka scan: per node, min edge leaving its component.
//     One wave per node; component labels staged in LDS; float4 row loads;
//     u64 atomicMin into per-component best (GLOBAL_ATOMIC_MIN_U64).
// ---------------------------------------------------------------------------
__global__ __launch_bounds__(256) void boruvka_scan_kernel(float* __restrict__ ws) {
    const int matrix = blockIdx.y;
    const float* Dmat = ws + kOffD + (size_t)matrix * kNN;
    const int* comp   = (const int*)(ws + kOffComp) + (size_t)matrix * kN;
    unsigned long long* cbest =
        (unsigned long long*)(ws + kOffCbest) + (size_t)matrix * kN;

    __shared__ int scomp[kN];
    for (int t = threadIdx.x; t < kN; t += 256) scomp[t] = comp[t];
    __syncthreads();

    const int lane = threadIdx.x & 31;
    const int wid  = threadIdx.x >> 5;
    const int i    = blockIdx.x * 8 + wid;         // node owned by this wave
    const int ci   = scomp[i];

    const float* row = Dmat + (size_t)i * kN;
    unsigned long long best = kSent;
    #pragma unroll 4
    for (int it = 0; it < kN / 128; ++it) {
        int j0 = it * 128 + lane * 4;
        v4f w = *(const v4f*)(row + j0);
        #pragma unroll
        for (int t = 0; t < 4; ++t) {
            int j = j0 + t;
            if (scomp[j] != ci) {
                unsigned long long key = pack_key(w[t], i, j);
                if (key < best) best = key;
            }
        }
    }
    #pragma unroll
    for (int off = 16; off > 0; off >>= 1) {
        unsigned long long o = shfl_xor_u64(best, off);
        if (o < best) best = o;
    }
    if (lane == 0 && best != kSent) atomicMin(&cbest[ci], best);
}

// ---------------------------------------------------------------------------
// 3b) Boruvka merge: each component root attaches along its best edge.
//     Mutual (a<->b) pairs: smaller label records the death & attaches;
//     larger label stays root. Guarantees each edge recorded exactly once.
// ---------------------------------------------------------------------------
__global__ __launch_bounds__(256) void boruvka_merge_kernel(float* __restrict__ ws) {
    int id = blockIdx.x * blockDim.x + threadIdx.x;   // 0 .. 2*kN-1
    int m = id >> 12;
    int a = id & (kN - 1);
    const int* comp = (const int*)(ws + kOffComp) + (size_t)m * kN;
    int* parent     = (int*)(ws + kOffParent) + (size_t)m * kN;
    const unsigned long long* cbest =
        (const unsigned long long*)(ws + kOffCbest) + (size_t)m * kN;
    float* deaths = ws + kOffDeaths + (size_t)m * kN;
    int* cnt      = (int*)(ws + kOffCnt) + m;

    unsigned long long e = cbest[a];
    if (e == kSent) return;                        // not an active root

    int j = (int)(e & 0xFFFu);
    int b = comp[j];                               // target component (!= a)

    bool mutual = false;
    unsigned long long eb = cbest[b];
    if (eb != kSent) {
        int jb = (int)(eb & 0xFFFu);
        mutual = (comp[jb] == a);
    }
    if (mutual && a > b) return;                   // partner records this edge

    parent[a] = b;
    int idx = atomicAdd(cnt, 1);
    deaths[idx] = __uint_as_float((unsigned int)(e >> 32));
}

// ---------------------------------------------------------------------------
// 3c) pointer-jump: flatten labels to current roots; reset cbest for next
//     round. Chains only traverse this round's roots, so chase terminates.
// ---------------------------------------------------------------------------
__global__ __launch_bounds__(256) void boruvka_jump_kernel(float* __restrict__ ws) {
    int id = blockIdx.x * blockDim.x + threadIdx.x;   // 0 .. 2*kN-1
    int m = id >> 12;
    int i = id & (kN - 1);
    int* comp         = (int*)(ws + kOffComp) + (size_t)m * kN;
    const int* parent = (const int*)(ws + kOffParent) + (size_t)m * kN;

    int p = comp[i];
    #pragma unroll 1
    for (int s = 0; s < kN; ++s) {
        int q = parent[p];
        if (q == p) break;
        p = q;
    }
    comp[i] = p;

    ((unsigned long long*)(ws + kOffCbest))[id] = kSent;   // reset for next round
}

// ---------------------------------------------------------------------------
// 4) sort deaths ascending (bitonic, in LDS), one block per matrix
// ---------------------------------------------------------------------------
__global__ __launch_bounds__(1024) void sort_kernel(float* __restrict__ ws) {
    const int matrix = blockIdx.x;
    float* deaths = ws + kOffDeaths + (size_t)matrix * kN;

    __shared__ float sd[kN];
    const int tid = threadIdx.x;
    for (int r = 0; r < 4; ++r) {
        int i = tid + r * 1024;
        sd[i] = (i < kN - 1) ? deaths[i] : kBIG;   // pad slot 4095
    }
    __syncthreads();

    for (unsigned k = 2; k <= (unsigned)kN; k <<= 1) {
        for (unsigned j = k >> 1; j > 0; j >>= 1) {
            for (unsigned t = tid; t < (unsigned)kN / 2; t += 1024) {
                unsigned i = 2u * j * (t / j) + (t % j);
                unsigned p = i + j;
                bool up = ((i & k) == 0);
                float a = sd[i], b = sd[p];
                if ((a > b) == up) { sd[i] = b; sd[p] = a; }
            }
            __syncthreads();
        }
    }

    for (int r = 0; r < 4; ++r) {
        int i = tid + r * 1024;
        deaths[i] = sd[i];
    }
}

// ---------------------------------------------------------------------------
// 5) representation MSE partial sums -> fp32 atomic accumulator
// ---------------------------------------------------------------------------
__global__ __launch_bounds__(256) void mse_kernel(
    const float* __restrict__ a, const float* __restrict__ b,
    float* __restrict__ ws, int n) {
    __shared__ float s[256];
    int base = blockIdx.x * 1024 + threadIdx.x;
    float sum = 0.0f;
    #pragma unroll
    for (int r = 0; r < 4; ++r) {
        int i = base + r * 256;
        if (i < n) { float d = a[i] - b[i]; sum += d * d; }
    }
    s[threadIdx.x] = sum;
    __syncthreads();
    for (int off = 128; off > 0; off >>= 1) {
        if (threadIdx.x < off) s[threadIdx.x] += s[threadIdx.x + off];
        __syncthreads();
    }
    if (threadIdx.x == 0) atomicAdd(ws + kOffAccum, s[0]);
}

// ---------------------------------------------------------------------------
// 6) combine: topo MSE over paired sorted deaths + final weighted sum
// ---------------------------------------------------------------------------
__global__ __launch_bounds__(1024) void finalize_kernel(
    const float* __restrict__ ws, float* __restrict__ out) {
    __shared__ float s[1024];
    const float* ds = ws + kOffDeaths;
    const float* dt = ws + kOffDeaths + kN;
    float sum = 0.0f;
    for (int i = threadIdx.x; i < kN - 1; i += 1024) {
        float d = ds[i] - dt[i];
        sum += d * d;
    }
    s[threadIdx.x] = sum;
    __syncthreads();
    for (int off = 512; off > 0; off >>= 1) {
        if (threadIdx.x < off) s[threadIdx.x] += s[threadIdx.x + off];
        __syncthreads();
    }
    if (threadIdx.x == 0) {
        // diagram is [N-1, 2] with births all zero -> divide by 2*(N-1)
        float topo = s[0] / (2.0f * (float)(kN - 1));
        float repr = ws[kOffAccum] / (float)((size_t)kN * kD);
        out[0] = 0.5f * repr + 0.5f * topo;
        out[1] = repr;
        out[2] = topo;
    }
}

// ---------------------------------------------------------------------------
extern "C" void kernel_launch(void* const* d_in, const int* in_sizes, int n_in,
                              void* d_out, int out_size, void* d_ws, size_t ws_size,
                              hipStream_t stream) {
    const float* Xs = (const float*)d_in[0];   // student_repr [4096,128] f32
    const float* Xt = (const float*)d_in[1];   // teacher_repr [4096,128] f32
    // d_in[2], d_in[3] (edges) are unused by the reference.
    float* ws  = (float*)d_ws;
    float* out = (float*)d_out;

    // reset all per-call state (harness never re-zeros ws between replays)
    init_kernel<<<(2 * kN) / 256, 256, 0, stream>>>(ws);

    // row squared norms: 2*4096 rows
    sqnorm_kernel<<<(2 * kN) / 256, 256, 0, stream>>>(Xs, Xt, ws);

    // WMMA GEMM + distance epilogue: 256x256 tiles, 8 waves/block, 2 matrices
    dim3 gg((kN / 16) * (kN / 16) / 8, 2, 1);
    gemm_dist_kernel<<<gg, 256, 0, stream>>>(Xs, Xt, ws);

    // Boruvka MST: 12 fully-parallel rounds (later rounds are no-ops once
    // everything is merged; death count ends at exactly N-1 per matrix)
    dim3 gs(kN / 8, 2, 1);                     // one wave per node
    for (int r = 0; r < kRounds; ++r) {
        boruvka_scan_kernel<<<gs, 256, 0, stream>>>(ws);
        boruvka_merge_kernel<<<(2 * kN) / 256, 256, 0, stream>>>(ws);
        boruvka_jump_kernel<<<(2 * kN) / 256, 256, 0, stream>>>(ws);
    }

    // sort deaths (one block per matrix)
    sort_kernel<<<2, 1024, 0, stream>>>(ws);

    // representation MSE: 4096*128 elements, 4/thread
    int n = kN * kD;
    mse_kernel<<<(n + 1023) / 1024, 256, 0, stream>>>(Xs, Xt, ws, n);

    finalize_kernel<<<1, 1024, 0, stream>>>(ws, out);
}